// Emb_ROSA_47665547051799
// MI455X (gfx1250) — compile-verified
//
#include <hip/hip_runtime.h>
#include <stdint.h>

#define B_ 8
#define T_ 2048
#define C_ 1024

typedef unsigned int u32;
typedef u32 __attribute__((ext_vector_type(4))) uv4;
typedef int __attribute__((ext_vector_type(4))) iv4;
typedef int __attribute__((ext_vector_type(8))) iv8;

// ---------------- Kernel: embedding gather via TDM (placed FIRST so the
// ---------------- disasm snippet shows the tensor_load/store encodings) ----
#if __has_builtin(__builtin_amdgcn_tensor_load_to_lds) && \
    __has_builtin(__builtin_amdgcn_tensor_store_from_lds)
#define USE_TDM 1
#else
#define USE_TDM 0
#warning "CDNA5 TDM builtins unavailable -- falling back to float4 vector copy"
#endif

__global__ __launch_bounds__(32) void k_gather(const int* __restrict__ idx,
                                               const float* __restrict__ emb,
                                               const u32* __restrict__ keys,
                                               float* __restrict__ out) {
  const int bi = (int)blockIdx.x;   // flat (b, i), one wave per output row
  const int b  = bi >> 11;          // T_ == 2048
  const u32 key = keys[bi];
  float* dst = out + (size_t)bi * C_;

  if (key == 0u) {                  // no earlier suffix match -> zeros
    float4 z = {0.f, 0.f, 0.f, 0.f};
    float4* d4 = (float4*)dst;
    for (int k = (int)threadIdx.x; k < C_ / 4; k += 32) d4[k] = z;
    return;
  }

  const int j   = (int)(key & 0xFFFFu);
  const int tok = idx[b * T_ + j + 1];
  const float* src = emb + (size_t)tok * C_;

#if USE_TDM
  __shared__ float lbuf[C_];
  // low 32 bits of the generic address == LDS byte offset (aperture in [63:32])
  const u32 lo = (u32)(uintptr_t)(void*)&lbuf[0];

  // D# group 1 (ISA 8.4): data_size=2 (4B); tensor_dim0=tile_dim0=1024;
  // tensor_dim1=tile_dim1=1; strides=1024 elements; no pad/iterate/multicast.
  iv8 g1;
  g1[0] = (int)(2u << 16);      // data_size in bits[17:16]
  g1[1] = (int)(1024u << 16);   // tensor_dim0[15:0] -> bits[63:48]
  g1[2] = (int)(1u << 16);      // tensor_dim1[15:0] -> bits[95:80]
  g1[3] = (int)(1024u << 16);   // tile_dim0 -> bits[127:112]
  g1[4] = 1;                    // tile_dim1 = 1
  g1[5] = 1024;                 // tensor_dim0_stride[31:0]
  g1[6] = (int)(1024u << 16);   // tensor_dim1_stride[15:0]
  g1[7] = 0;
  iv4 z4 = {0, 0, 0, 0};

  const unsigned long long ga = (unsigned long long)(uintptr_t)src;
  const unsigned long long da = (unsigned long long)(uintptr_t)dst;

  // D# group 0 (ISA 8.3): count=1, lds_addr, global_addr[56:0], type=2
  uv4 g0l, g0s;
  g0l[0] = 1u;
  g0l[1] = lo;
  g0l[2] = (u32)ga;
  g0l[3] = (u32)((ga >> 32) & 0x1FFFFFFu) | (2u << 30);
  g0s[0] = 1u;
  g0s[1] = lo;
  g0s[2] = (u32)da;
  g0s[3] = (u32)((da >> 32) & 0x1FFFFFFu) | (2u << 30);

#if defined(__clang_major__) && (__clang_major__ >= 23)
  iv8 z8 = {0, 0, 0, 0, 0, 0, 0, 0};
  __builtin_amdgcn_tensor_load_to_lds(g0l, g1, z4, z4, z8, 0);
  __builtin_amdgcn_s_wait_tensorcnt(0);    // load complete in LDS
  __builtin_amdgcn_tensor_store_from_lds(g0s, g1, z4, z4, z8, 0);
  __builtin_amdgcn_s_wait_tensorcnt(0);    // store drained before LDS reuse
#else
  __builtin_amdgcn_tensor_load_to_lds(g0l, g1, z4, z4, 0);
  __builtin_amdgcn_s_wait_tensorcnt(0);
  __builtin_amdgcn_tensor_store_from_lds(g0s, g1, z4, z4, 0);
  __builtin_amdgcn_s_wait_tensorcnt(0);
#endif
#else
  const float4* s4 = (const float4*)src;
  float4* d4 = (float4*)dst;
  for (int k = (int)threadIdx.x; k < C_ / 4; k += 32) d4[k] = s4[k];
#endif
}

// ---------------- Kernel: diagonal suffix-match DP ------------------------
// m(i,j) = len of common suffix of prefixes ending at i and j. Along a
// diagonal d=i-j it is the running count of consecutive equal token pairs.
// keys[b][i] = max over j<i of ((m<<16)|j): longest match, most-recent j.
__global__ __launch_bounds__(256) void k_dp(const int* __restrict__ idx,
                                            u32* __restrict__ keys) {
  __shared__ int row[T_];
  const int b = blockIdx.y;
  const int* x = idx + b * T_;
  for (int t = (int)threadIdx.x; t < T_; t += 256) row[t] = x[t];
  __syncthreads();

  const int d = (int)(blockIdx.x * 256 + threadIdx.x) + 1;  // diagonal 1..T_-1
  if (d >= T_) return;
  u32* krow = keys + b * T_;
  u32 run = 0;
  for (int t = 0; t + d < T_; ++t) {
    if (row[t + d] == row[t]) {
      ++run;  // match of length `run` ending at i=t+d, earlier end j=t
      atomicMax(&krow[t + d], (run << 16) | (u32)t);
    } else {
      run = 0;
    }
  }
}

// ---------------- Kernel: zero the packed match keys ----------------------
__global__ void k_zero_keys(u32* __restrict__ keys, int n) {
  int i = blockIdx.x * blockDim.x + threadIdx.x;
  if (i < n) keys[i] = 0u;
}

// --------------------------------------------------------------------------
extern "C" void kernel_launch(void* const* d_in, const int* in_sizes, int n_in,
                              void* d_out, int out_size, void* d_ws, size_t ws_size,
                              hipStream_t stream) {
  (void)in_sizes; (void)n_in; (void)out_size; (void)ws_size;
  const int*   idx = (const int*)d_in[0];     // [B, T] tokens
  const float* emb = (const float*)d_in[1];   // [V, C] f32
  float*       out = (float*)d_out;           // [B, T, C] f32
  u32*         keys = (u32*)d_ws;             // [B*T] packed (m<<16)|j

  const int n = B_ * T_;
  k_zero_keys<<<(n + 255) / 256, 256, 0, stream>>>(keys, n);
  k_dp<<<dim3(T_ / 256, B_), 256, 0, stream>>>(idx, keys);
  k_gather<<<n, 32, 0, stream>>>(idx, emb, keys, out);
}